// FastweightMemory_16192026706529
// MI455X (gfx1250) — compile-verified
//
#include <hip/hip_runtime.h>
#include <hip/hip_bf16.h>
#include <math.h>

typedef __attribute__((ext_vector_type(2))) float v2f;
typedef __attribute__((ext_vector_type(8))) float v8f;

#define B_ 4
#define H_ 16
#define N_ 8192
#define D_ 64
#define E_ 64
#define CHUNKS 32
#define WAVES 4
#define TPB 128
#define EPS_ 1e-10f

// LDS layout strides (padding chosen to break bank conflicts; see analysis)
#define S2 65            // row-pair stride (v2f units) for kv / k / v' tiles
#define SACC 66          // dword stride for kv_acc rows
#define SVR 68           // dword stride for raw staging tiles (16B-aligned rows)

// stop cross-phase instruction motion (keeps live ranges small); async loads still overlap
#define PHASE_FENCE() __builtin_amdgcn_sched_barrier(0)

__device__ __forceinline__ float elu1(float x) {
  // elu(x)+1 == x+1 (x>0) else exp(x)
  return x > 0.f ? x + 1.f : __expf(x);
}

__global__ void fw_init_kernel(const float* __restrict__ past_kv,
                               const float* __restrict__ past_norm,
                               float* __restrict__ new_kv,
                               float* __restrict__ new_norm) {
  int i = blockIdx.x * blockDim.x + threadIdx.x;
  if (i < B_ * H_ * D_ * E_) new_kv[i] = past_kv[i];
  if (i < B_ * H_ * D_)      new_norm[i] = past_norm[i];
}

__global__ __launch_bounds__(TPB, 4) void fw_main_kernel(
    const float* __restrict__ keys,
    const float* __restrict__ values,
    const float* __restrict__ queries,
    const float* __restrict__ out,
    const float* __restrict__ past_kv,
    const float* __restrict__ past_norm,
    const float* __restrict__ head_gates,
    float* __restrict__ out_g,
    float* __restrict__ new_kv,
    float* __restrict__ new_norm)
{
  extern __shared__ float smem[];
  // kv2_s: past_kv as 32 row-pairs x 64 cols of float2 {kv[2m][c], kv[2m+1][c]}, stride S2
  v2f*   kv2_s    = (v2f*)smem;                 // 32*S2 v2f   = 4160 floats
  float* kv_acc   = smem + 2 * 32 * S2;         // 64*SACC     = 4224 floats
  float* norm_s   = kv_acc + 64 * SACC;         // 64
  float* norm_acc = norm_s + 64;                // 64
  float* stage    = norm_acc + 64;              // WAVES * (2*2*8*S2 + 2*16*SVR) floats

  const int tid  = threadIdx.x;
  const int wave = tid >> 5;
  const int lane = tid & 31;
  const int lo   = lane & 15;   // row / col index within 16
  const int hi   = lane >> 4;   // half select

  const int bh = blockIdx.x;           // b*H + h
  const int h  = bh & (H_ - 1);
  const int chunk = blockIdx.y;

  const size_t base_nd = (size_t)bh * N_ * D_;
  const float* keys_bh    = keys    + base_nd;
  const float* values_bh  = values  + base_nd;
  const float* queries_bh = queries + base_nd;
  const float* out_bh     = out     + base_nd;
  float*       outg_bh    = out_g   + base_nd;
  const float* kv_bh   = past_kv   + (size_t)bh * D_ * E_;
  const float* norm_bh = past_norm + (size_t)bh * D_;

  const float gate  = 1.f / (1.f + __expf(-head_gates[h]));
  const float gatec = 1.f - gate;

  // fill paired-row past_kv tile; zero accumulators
  for (int i = tid; i < 32 * 64; i += TPB) {
    const int m = i >> 6, c = i & 63;
    v2f p;
    p.x = kv_bh[(2 * m) * E_ + c];
    p.y = kv_bh[(2 * m + 1) * E_ + c];
    kv2_s[m * S2 + c] = p;
  }
  for (int i = tid; i < 64 * SACC; i += TPB) kv_acc[i] = 0.f;
  if (tid < D_) { norm_s[tid] = norm_bh[tid]; norm_acc[tid] = 0.f; }
  __syncthreads();

  // per-wave staging: k/v' paired-row tiles + two raw tiles (keys|out, values|queries)
  const int wave_floats = 2 * 2 * 8 * S2 + 2 * 16 * SVR;     // 2080 + 2176
  float* wbase = stage + wave * wave_floats;
  v2f* k2_s = (v2f*)wbase;
  v2f* v2_s = k2_s + 8 * S2;
  float* k2f  = (float*)k2_s;
  float* v2f_ = (float*)v2_s;
  float* kr_s = wbase + 2 * 2 * 8 * S2;          // raw keys, later raw out
  float* vr_s = kr_s + 16 * SVR;                 // raw values, later raw queries

  const int tiles = (N_ / CHUNKS) / 16;          // 16 tiles of 16 rows
  const int row_chunk0 = chunk * (N_ / CHUNKS);

  for (int t = wave; t < tiles; t += WAVES) {
    const int r0 = row_chunk0 + t * 16;

    // ---- bulk-stage keys and values tiles (16 x b128 per wave, fully pipelined)
    #pragma unroll
    for (int i = 0; i < 8; ++i) {
      const int row = 2 * i + hi;
      const float4 kk = *(const float4*)(keys_bh   + (size_t)(r0 + row) * D_ + 4 * lo);
      const float4 vv = *(const float4*)(values_bh + (size_t)(r0 + row) * E_ + 4 * lo);
      *(float4*)(kr_s + row * SVR + 4 * lo) = kk;
      *(float4*)(vr_s + row * SVR + 4 * lo) = vv;
    }
    PHASE_FENCE();

    // ================= K path (fused: LDS read -> elu -> denom -> stage -> 4 WMMAs) ====
    v8f cacc[4] = {};
    float dpart = 0.f;
    #pragma unroll
    for (int kb = 0; kb < 16; ++kb) {
      const int c = 4 * kb + 2 * hi;
      const v2f kraw = *(const v2f*)(kr_s + lo * SVR + c);
      const float e0 = elu1(kraw.x);
      const float e1 = elu1(kraw.y);
      dpart += e0 * norm_s[c] + e1 * norm_s[c + 1];
      const int base = (lo >> 1) * (2 * S2) + (lo & 1);
      k2f[base + 2 * c]     = e0;
      k2f[base + 2 * c + 2] = e1;
      v2f a; a.x = e0; a.y = e1;
      const int mrow = (2 * kb + hi) * S2 + lo;
      #pragma unroll
      for (int et = 0; et < 4; ++et) {
        const v2f b = kv2_s[mrow + 16 * et];
        cacc[et] = __builtin_amdgcn_wmma_f32_16x16x4_f32(
            false, a, false, b, (short)0, cacc[et], false, false);
      }
    }
    PHASE_FENCE();

    // single reciprocal per lane; broadcast the 8 per-row reciprocals for the D-layout
    const float denom = fmaxf(dpart + __shfl_xor(dpart, 16, 32), EPS_);
    const float rden  = 1.f / denom;
    float rdj[8];
    #pragma unroll
    for (int j = 0; j < 8; ++j) rdj[j] = __shfl(rden, j + 8 * hi, 32);

    // v' = values - numer * (1/denom); values from LDS staging, v' to paired-row tile
    #pragma unroll
    for (int et = 0; et < 4; ++et) {
      #pragma unroll
      for (int j = 0; j < 8; ++j) {
        const int rloc = j + 8 * hi;                       // D-layout row
        const float val = vr_s[rloc * SVR + 16 * et + lo];
        v2f_[(rloc >> 1) * (2 * S2) + 2 * (16 * et + lo) + (rloc & 1)]
            = val - cacc[et][j] * rdj[j];
      }
    }
    PHASE_FENCE();

    // ---- re-stage: queries -> vr_s, out -> kr_s (loads stay outstanding across phases)
    #pragma unroll
    for (int i = 0; i < 8; ++i) {
      const int row = 2 * i + hi;
      const float4 qq = *(const float4*)(queries_bh + (size_t)(r0 + row) * D_ + 4 * lo);
      const float4 oo = *(const float4*)(out_bh     + (size_t)(r0 + row) * E_ + 4 * lo);
      *(float4*)(vr_s + row * SVR + 4 * lo) = qq;
      *(float4*)(kr_s + row * SVR + 4 * lo) = oo;
    }
    PHASE_FENCE();

    // new_norm partial: column sums of k tile (pairs already adjacent)
    {
      float s0 = 0.f, s1 = 0.f;
      #pragma unroll
      for (int m = 0; m < 8; ++m) {
        const v2f p0 = k2_s[m * S2 + 2 * lane];
        const v2f p1 = k2_s[m * S2 + 2 * lane + 1];
        s0 += p0.x + p0.y;
        s1 += p1.x + p1.y;
      }
      atomicAdd(&norm_acc[2 * lane], s0);
      atomicAdd(&norm_acc[2 * lane + 1], s1);
    }

    // kv_acc += k^T v'  (16 output tiles of 16x16, K=16 rows as 4 steps of 4)
    #pragma unroll
    for (int dt = 0; dt < 4; ++dt) {
      #pragma unroll
      for (int et = 0; et < 4; ++et) {
        v8f acc = {};
        #pragma unroll
        for (int rb = 0; rb < 4; ++rb) {
          const int mrow = (2 * rb + hi) * S2;
          const v2f af = k2_s[mrow + 16 * dt + lo];
          const v2f bf = v2_s[mrow + 16 * et + lo];
          acc = __builtin_amdgcn_wmma_f32_16x16x4_f32(
              false, af, false, bf, (short)0, acc, false, false);
        }
        #pragma unroll
        for (int j = 0; j < 8; ++j)
          atomicAdd(&kv_acc[(16 * dt + j + 8 * hi) * SACC + 16 * et + lo], acc[j]);
      }
    }
    PHASE_FENCE();

    // ================= Q path (fused) =================
    v8f qacc[4] = {};
    float qdp = 0.f;
    #pragma unroll
    for (int kb = 0; kb < 16; ++kb) {
      const int c = 4 * kb + 2 * hi;
      const v2f qraw = *(const v2f*)(vr_s + lo * SVR + c);
      const float e0 = elu1(qraw.x);
      const float e1 = elu1(qraw.y);
      qdp += e0 * norm_s[c] + e1 * norm_s[c + 1];
      v2f a; a.x = e0; a.y = e1;
      const int mrow = (2 * kb + hi) * S2 + lo;
      #pragma unroll
      for (int et = 0; et < 4; ++et) {
        const v2f b = kv2_s[mrow + 16 * et];
        qacc[et] = __builtin_amdgcn_wmma_f32_16x16x4_f32(
            false, a, false, b, (short)0, qacc[et], false, false);
      }
    }
    PHASE_FENCE();

    const float qden = fmaxf(qdp + __shfl_xor(qdp, 16, 32), EPS_);
    const float qrden = 1.f / qden;
    float qrdj[8];
    #pragma unroll
    for (int j = 0; j < 8; ++j) qrdj[j] = __shfl(qrden, j + 8 * hi, 32);

    // gate blend; 'out' comes from LDS staging, stores are fire-and-forget
    #pragma unroll
    for (int et = 0; et < 4; ++et) {
      #pragma unroll
      for (int j = 0; j < 8; ++j) {
        const int rloc = j + 8 * hi;
        const float oo = kr_s[rloc * SVR + 16 * et + lo];
        const float mem = qacc[et][j] * qrdj[j];
        outg_bh[(size_t)(r0 + rloc) * E_ + 16 * et + lo] = oo * gate + mem * gatec;
      }
    }
    PHASE_FENCE();
  }

  // flush per-block accumulators into global outputs (init kernel seeded past_*)
  __syncthreads();
  float* nkv_bh = new_kv + (size_t)bh * D_ * E_;
  for (int i = tid; i < D_ * E_; i += TPB) {
    const int r = i >> 6, c = i & 63;
    atomicAdd(&nkv_bh[i], kv_acc[r * SACC + c]);
  }
  if (tid < D_) atomicAdd(&new_norm[(size_t)bh * D_ + tid], norm_acc[tid]);
}

extern "C" void kernel_launch(void* const* d_in, const int* in_sizes, int n_in,
                              void* d_out, int out_size, void* d_ws, size_t ws_size,
                              hipStream_t stream) {
  const float* keys       = (const float*)d_in[0];
  const float* values     = (const float*)d_in[1];
  const float* queries    = (const float*)d_in[2];
  const float* out        = (const float*)d_in[3];
  const float* past_kv    = (const float*)d_in[4];
  const float* past_norm  = (const float*)d_in[5];
  const float* head_gates = (const float*)d_in[6];

  float* outp     = (float*)d_out;
  float* out_g    = outp;                                      // [B,H,N,E]
  float* new_kv   = outp + (size_t)B_ * H_ * N_ * E_;          // [B,H,D,E]
  float* new_norm = new_kv + (size_t)B_ * H_ * D_ * E_;        // [B,H,D]

  // seed new_kv/new_norm with past values
  const int init_n = B_ * H_ * D_ * E_;
  fw_init_kernel<<<(init_n + 255) / 256, 256, 0, stream>>>(past_kv, past_norm, new_kv, new_norm);

  dim3 grid(B_ * H_, CHUNKS);
  // floats: kv2 (2*32*S2) + kv_acc (64*SACC) + norms (128) + stage (WAVES*(2*2*8*S2 + 2*16*SVR))
  const size_t smem_floats = (size_t)(2 * 32 * S2) + 64 * SACC + 128
                           + (size_t)WAVES * (2 * 2 * 8 * S2 + 2 * 16 * SVR);
  fw_main_kernel<<<grid, TPB, smem_floats * sizeof(float), stream>>>(
      keys, values, queries, out, past_kv, past_norm, head_gates,
      out_g, new_kv, new_norm);
}